// QuantumDARTSModel_76897094468165
// MI455X (gfx1250) — compile-verified
//
#include <hip/hip_runtime.h>
#include <hip/hip_bf16.h>

// ---------------------------------------------------------------------------
// QuantumDARTS forward on MI455X (gfx1250).
// One workgroup per batch element; the full 2^15 complex-f32 statevector
// (256 KB) lives in LDS (320 KB/WGP). Single-qubit gates are applied with
// V_WMMA_F32_16X16X4_F32: real 4x4 gate matrix expanded into a 16x4 A operand
// whose rows 0-7 produce re components and rows 8-15 produce im components,
// so the D write-back uses only constant vector indices (no cndmask ladders).
// CNOTs are in-place LDS swaps. Class probs via ds_add_f32. Zero HBM traffic
// for the state.
// ---------------------------------------------------------------------------

#define NQ       15
#define DIM      (1 << NQ)        // 32768 amplitudes
#define NPAIRS   (DIM >> 1)       // 16384 pairs per gate
#define LAYERS   4
#define KP       4
#define NG       3
#define NCLASSES 4
#define THREADS  1024
#define WAVES    (THREADS / 32)

typedef __attribute__((ext_vector_type(2))) float v2f;
typedef __attribute__((ext_vector_type(8))) float v8f;

// Real 4x4 matrix for RX/RY/RZ(theta) acting on (re0, re1, im0, im1):
// M = [[Ur, -Ui], [Ui, Ur]] with U = Ur + i*Ui.
__device__ __forceinline__ void build_M(float* sM, int g, float th) {
    float c = __cosf(0.5f * th);
    float s = __sinf(0.5f * th);
    float ur00, ur01, ur10, ur11, ui00, ui01, ui10, ui11;
    if (g == 0) {            // RX
        ur00 = c;  ur01 = 0.f; ur10 = 0.f; ur11 = c;
        ui00 = 0.f; ui01 = -s; ui10 = -s;  ui11 = 0.f;
    } else if (g == 1) {     // RY
        ur00 = c;  ur01 = -s;  ur10 = s;   ur11 = c;
        ui00 = 0.f; ui01 = 0.f; ui10 = 0.f; ui11 = 0.f;
    } else {                 // RZ
        ur00 = c;  ur01 = 0.f; ur10 = 0.f; ur11 = c;
        ui00 = -s; ui01 = 0.f; ui10 = 0.f; ui11 = s;
    }
    sM[0]  = ur00; sM[1]  = ur01; sM[2]  = -ui00; sM[3]  = -ui01;
    sM[4]  = ur10; sM[5]  = ur11; sM[6]  = -ui10; sM[7]  = -ui11;
    sM[8]  = ui00; sM[9]  = ui01; sM[10] =  ur00; sM[11] =  ur01;
    sM[12] = ui10; sM[13] = ui11; sM[14] =  ur10; sM[15] =  ur11;
}

__global__ __launch_bounds__(THREADS)
void qdarts_sim_kernel(const float* __restrict__ x,
                       const float* __restrict__ P,
                       const float* __restrict__ Q,
                       const float* __restrict__ rot,
                       const float* __restrict__ gum,
                       float* __restrict__ out)
{
    extern __shared__ float lds[];          // re[DIM] | im[DIM]  (256 KB)
    float* re = lds;
    float* im = lds + DIM;
    __shared__ float sM[16];
    __shared__ float sCls[NCLASSES];

    const int tid  = threadIdx.x;
    const int bid  = blockIdx.x;            // batch element
    const int lane = tid & 31;
    const int wave = tid >> 5;
    const int ncol = lane & 15;             // WMMA column (pair slot / A row)
    const int half = lane >> 4;             // lane half (K split)

    // A-row remap: rows 0-7 -> M4[m&3] (re rows), rows 8-15 -> M4[(m&3)^2]
    // (im rows), so D's lane halves hold re / im at constant extract indices.
    const int srow = (ncol & 3) ^ ((ncol >> 2) & 2);

    // Each lane-half works on its own component plane of the state.
    float* const plane = half ? im : re;

    // ---- init |0...0> ----
    for (int i = tid; i < DIM; i += THREADS) { re[i] = 0.f; im[i] = 0.f; }
    if (tid == 0) {
        re[0] = 1.f;
        for (int c = 0; c < NCLASSES; ++c) sCls[c] = 0.f;
    }
    __syncthreads();

    // ---- WMMA single-qubit gate application ----
    // A (16x4 f32): VGPR0 -> K=0 (lanes 0-15) / K=2 (lanes 16-31),
    //               VGPR1 -> K=1 / K=3   (per CDNA5 ISA A-layout).
    // B (4x16 f32): symmetric layout assumed: half0 supplies rows K=0,1
    //               (re0,re1), half1 supplies K=2,3 (im0,im1) == plane loads.
    // D: d[0] = re0' (half0 lanes) / im0' (half1 lanes, via row remap),
    //    d[1] = re1' / im1'  -> all-lane stores with constant indices.
    auto apply_gate = [&](int bitpos) {
        const int s = 1 << bitpos;
        v2f a;
        a.x = sM[srow * 4 + (half ? 2 : 0)];
        a.y = sM[srow * 4 + (half ? 3 : 1)];
        #pragma unroll 4
        for (int it = 0; it < NPAIRS / (WAVES * 16); ++it) {
            const int t  = (it * WAVES + wave) * 16 + ncol;
            const int i0 = ((t >> bitpos) << (bitpos + 1)) | (t & (s - 1));
            const int i1 = i0 | s;
            v2f bm;
            bm.x = plane[i0];
            bm.y = plane[i1];
            v8f c = {};
            v8f d = __builtin_amdgcn_wmma_f32_16x16x4_f32(
                false, a, false, bm, (short)0, c, false, false);
            plane[i0] = d[0];
            plane[i1] = d[1];
        }
    };

    // ---- angle encoding: RY(x[b,q]) on qubit q ----
    for (int q = 0; q < NQ; ++q) {
        if (tid == 0) build_M(sM, 1, x[bid * NQ + q]);
        __syncthreads();
        apply_gate(NQ - 1 - q);
        __syncthreads();
    }

    // ---- layers: selected rotation per qubit, then CNOT ring ----
    for (int l = 0; l < LAYERS; ++l) {
        for (int q = 0; q < NQ; ++q) {
            if (tid == 0) {
                const int lq = l * NQ + q;
                // hard Gumbel-softmax forward == argmax(logits + noise)
                float best = -3.4e38f;
                int   gsel = 0;
                for (int g = 0; g < NG; ++g) {
                    float acc = gum[lq * NG + g];
                    for (int k = 0; k < KP; ++k)
                        acc += P[lq * KP + k] * Q[(lq * KP + k) * NG + g];
                    if (acc > best) { best = acc; gsel = g; }
                }
                build_M(sM, gsel, rot[lq]);
            }
            __syncthreads();
            apply_gate(NQ - 1 - q);
            __syncthreads();
        }
        // CNOT ring: control q, target (q+1)%NQ  (in-place LDS swaps)
        for (int q = 0; q < NQ; ++q) {
            const int cpos = NQ - 1 - q;
            const int tpos = NQ - 1 - ((q + 1) % NQ);
            const int lo   = cpos < tpos ? cpos : tpos;
            const int hi   = cpos < tpos ? tpos : cpos;
            for (int j = tid; j < (DIM >> 2); j += THREADS) {
                int tmp = ((j >> lo) << (lo + 1)) | (j & ((1 << lo) - 1));
                int idx = ((tmp >> hi) << (hi + 1)) | (tmp & ((1 << hi) - 1));
                const int ia = idx | (1 << cpos);      // control bit = 1
                const int ib = ia  | (1 << tpos);      // flip target
                float tr = re[ia]; re[ia] = re[ib]; re[ib] = tr;
                float ti = im[ia]; im[ia] = im[ib]; im[ib] = ti;
            }
            __syncthreads();
        }
    }

    // ---- class probabilities: sum |amp|^2 over contiguous blocks of 8192 ----
    {
        const int chunk = DIM / THREADS;          // 32 (class-aligned)
        const int base  = tid * chunk;
        float acc = 0.f;
        #pragma unroll
        for (int i = 0; i < chunk; ++i) {
            const float r = re[base + i];
            const float m = im[base + i];
            acc += r * r + m * m;
        }
        atomicAdd(&sCls[base >> (NQ - 2)], acc);  // ds_add_f32
    }
    __syncthreads();
    if (tid < NCLASSES) out[bid * NCLASSES + tid] = sCls[tid];
}

extern "C" void kernel_launch(void* const* d_in, const int* in_sizes, int n_in,
                              void* d_out, int out_size, void* d_ws, size_t ws_size,
                              hipStream_t stream) {
    const float* x   = (const float*)d_in[0];   // [64, 15]
    const float* P   = (const float*)d_in[1];   // [4, 15, 1, 4]
    const float* Q   = (const float*)d_in[2];   // [4, 15, 4, 3]
    const float* rot = (const float*)d_in[3];   // [60]
    const float* gum = (const float*)d_in[4];   // [4, 15, 3]
    float* out = (float*)d_out;                 // [64, 4]

    const size_t lds_bytes = size_t(2) * DIM * sizeof(float);   // 256 KB
    hipFuncSetAttribute((const void*)qdarts_sim_kernel,
                        hipFuncAttributeMaxDynamicSharedMemorySize,
                        (int)lds_bytes);
    qdarts_sim_kernel<<<64, THREADS, lds_bytes, stream>>>(x, P, Q, rot, gum, out);
}